// st_gcn_300647711184
// MI455X (gfx1250) — compile-verified
//
#include <hip/hip_runtime.h>
#include <hip/hip_bf16.h>

typedef float v2f __attribute__((ext_vector_type(2)));
typedef float v8f __attribute__((ext_vector_type(8)));

#define NB 16
#define TT 512
#define HH 64
#define CC 64
#define OO 64
#define KWIN 9
#define EE 256

// workspace layout in floats
#define ADJ_OFF  0                               // 16*64*64      = 65536
#define DIAG_OFF (ADJ_OFF + NB*HH*HH)            // 16*512*64     = 524288
#define WT_OFF   (DIAG_OFF + NB*TT*HH)           // 9*64*64       = 36864
#define Z_OFF    (WT_OFF + KWIN*OO*OO)           // 16*512*64*64  = 33554432

#define LDSTRIDE 68
#define BN_INV 0.9999950000375f   // 1/sqrt(1 + 1e-5)

// ---------------- phase 0: adjacency + weight transform ----------------

__global__ __launch_bounds__(256) void k_zero_adj(float* __restrict__ adj) {
    int i = blockIdx.x * 256 + threadIdx.x;
    if (i < NB * HH * HH) adj[i] = 0.0f;
}

__global__ __launch_bounds__(256) void k_scatter(const int* __restrict__ edges,
                                                 float* __restrict__ adj) {
    int i = blockIdx.x * 256 + threadIdx.x;      // i over N*E
    if (i >= NB * EE) return;
    int n = i >> 8;          // /E
    int e = i & (EE - 1);
    int e0 = edges[n * 2 * EE + e];
    int e1 = edges[n * 2 * EE + EE + e];
    atomicAdd(&adj[(n * HH + e0) * HH + e1], 1.0f);
}

__global__ __launch_bounds__(256) void k_wtrans(const float* __restrict__ W_tcn,
                                                float* __restrict__ Wt) {
    int i = blockIdx.x * 256 + threadIdx.x;      // Wt[k][o][o']
    if (i >= KWIN * OO * OO) return;
    int k   = i / (OO * OO);
    int rem = i - k * (OO * OO);
    int o   = rem >> 6;
    int op  = rem & 63;
    Wt[i] = W_tcn[(o * OO + op) * KWIN + k];
}

// ---------------- phase 1: attention -> diag ----------------
// diag[n,t,h] = (sum_k adj[n,h,k]*exp(lrelu(a1[k]+a2[h]) - m)) /
//               (sum_k           exp(lrelu(a1[k]+a2[h]) - m)) + 1

__global__ __launch_bounds__(64) void k_att_diag(const float* __restrict__ X,
                                                 const float* __restrict__ w1,
                                                 const float* __restrict__ b1p,
                                                 const float* __restrict__ w2,
                                                 const float* __restrict__ b2p,
                                                 const float* __restrict__ adj,
                                                 float* __restrict__ diag) {
    __shared__ float a1s[HH];
    int nt = blockIdx.x;
    int n  = nt >> 9;                    // / T
    int h  = threadIdx.x;

    const float* xr = X + ((long)nt * HH + h) * CC;
    float s1 = 0.f, s2 = 0.f;
    #pragma unroll 8
    for (int c = 0; c < CC; ++c) {
        float xv = xr[c];
        s1 += xv * w1[c];
        s2 += xv * w2[c];
    }
    float a2 = s2 + b2p[0];
    a1s[h] = s1 + b1p[0];
    __syncthreads();

    const float* ar = adj + (n * HH + h) * HH;
    float m = -1e30f;
    #pragma unroll 8
    for (int k = 0; k < HH; ++k) {
        float v = a1s[k] + a2;
        v = v > 0.f ? v : 0.01f * v;
        m = fmaxf(m, v);
    }
    float num = 0.f, den = 0.f;
    #pragma unroll 8
    for (int k = 0; k < HH; ++k) {
        float v = a1s[k] + a2;
        v = v > 0.f ? v : 0.01f * v;
        float e = __expf(v - m);
        den += e;
        num += ar[k] * e;
    }
    diag[(long)nt * HH + h] = num / den + 1.0f;
}

// ---------------- phase 2: support GEMM + diag scale + BN1 + ReLU -> z ----------------
// rows = flattened (n,t,h) [M], cols = O [N], K = C = 64. z[row][o], row-major.

__global__ __launch_bounds__(128) void k_support(const float* __restrict__ X,
                                                 const float* __restrict__ W_lin,
                                                 const float* __restrict__ b_lin,
                                                 const float* __restrict__ g1,
                                                 const float* __restrict__ be1,
                                                 const float* __restrict__ diag,
                                                 float* __restrict__ z) {
    __shared__ float Xs[64 * LDSTRIDE];   // A tile: [row][c]; reused to stage D
    __shared__ float Ws[64 * LDSTRIDE];   // B tile: natural [o][c] copy of W_lin
    const int tid = threadIdx.x;
    const long rowBase = (long)blockIdx.x * 64;

    // X rows contiguous: 64 rows x 64 c = 4096 floats
    const float4* Xg = (const float4*)(X + rowBase * CC);
    const float4* Wg = (const float4*)(W_lin);
    for (int i = tid; i < 1024; i += 128) {
        float4 v = Xg[i];
        int r = i >> 4;
        int c = (i & 15) * 4;
        float* p = &Xs[r * LDSTRIDE + c];
        p[0] = v.x; p[1] = v.y; p[2] = v.z; p[3] = v.w;
        float4 w = Wg[i];
        float* q = &Ws[r * LDSTRIDE + c];
        q[0] = w.x; q[1] = w.y; q[2] = w.z; q[3] = w.w;
    }
    __syncthreads();

    const int wave = tid >> 5;
    const int lane = tid & 31;
    const int l15  = lane & 15;
    const int hi   = lane >> 4;

    const v8f vzero = {0.f, 0.f, 0.f, 0.f, 0.f, 0.f, 0.f, 0.f};
    v8f acc[4];
    #pragma unroll
    for (int j = 0; j < 4; ++j) acc[j] = vzero;

    const int arow = (16 * wave + l15) * LDSTRIDE;
    #pragma unroll
    for (int kk = 0; kk < CC; kk += 4) {
        const int ka = kk + 2 * hi;                 // this lane's K pair
        v2f a = *(const v2f*)&Xs[arow + ka];        // A[m][ka..ka+1]
        #pragma unroll
        for (int j = 0; j < 4; ++j) {
            // B[ka][o] = W_lin[o][ka]: contiguous pair in natural layout
            v2f b = *(const v2f*)&Ws[(16 * j + l15) * LDSTRIDE + ka];
            acc[j] = __builtin_amdgcn_wmma_f32_16x16x4_f32(
                false, a, false, b, (short)0, acc[j], false, false);
        }
    }

    // epilogue: (+b_lin) * diag, BN1, ReLU; stage D into Xs[row_local][o]
    float bl[4], iv[4], bt[4];
    #pragma unroll
    for (int j = 0; j < 4; ++j) {
        int o = 16 * j + l15;
        bl[j] = b_lin[o];
        iv[j] = g1[o] * BN_INV;
        bt[j] = be1[o];
    }
    __syncthreads();   // all waves done reading tiles
    #pragma unroll
    for (int r = 0; r < 8; ++r) {
        int rl = 16 * wave + 8 * hi + r;            // local row (wave-private strip)
        float d = diag[rowBase + rl];
        #pragma unroll
        for (int j = 0; j < 4; ++j) {
            float v = (acc[j][r] + bl[j]) * d;
            v = v * iv[j] + bt[j];
            Xs[rl * LDSTRIDE + 16 * j + l15] = fmaxf(v, 0.f);
        }
    }
    __syncthreads();

    // coalesced float4 store of the 64x64 tile
    float4* zg = (float4*)(z + rowBase * OO);
    for (int i = tid; i < 1024; i += 128) {
        int r = i >> 4;
        int c = (i & 15) * 4;
        const float* p = &Xs[r * LDSTRIDE + c];
        float4 v = {p[0], p[1], p[2], p[3]};
        zg[i] = v;
    }
}

// ---------------- phase 3: temporal conv GEMM + BN2 + residual + ReLU ----------------
// y[n,o,t,h] = sum_k sum_o' Wt[k][o][o'] * z[n,t+k-4,h,o']   (zero pad in t)

__global__ __launch_bounds__(128) void k_conv(const float* __restrict__ z,
                                              const float* __restrict__ Wt,
                                              const float* __restrict__ btcn,
                                              const float* __restrict__ g2,
                                              const float* __restrict__ be2,
                                              const float* __restrict__ X,
                                              float* __restrict__ out) {
    __shared__ float Zs[64 * LDSTRIDE];   // B tile: [h][o']; reused to stage D
    __shared__ float Wk[64 * LDSTRIDE];   // A tile: [o][o']
    const int tid = threadIdx.x;
    const int nt  = blockIdx.x;
    const int n   = nt >> 9;
    const int t   = nt & 511;
    const int wave = tid >> 5;
    const int lane = tid & 31;
    const int l15  = lane & 15;
    const int hi   = lane >> 4;

    const v8f vzero = {0.f, 0.f, 0.f, 0.f, 0.f, 0.f, 0.f, 0.f};
    v8f acc[4];
    #pragma unroll
    for (int j = 0; j < 4; ++j) acc[j] = vzero;

    for (int k = 0; k < KWIN; ++k) {
        int tz = t + k - 4;
        if (tz < 0 || tz >= TT) continue;   // uniform across block
        __syncthreads();                    // protect LDS reuse across k
        const float4* Zg = (const float4*)(z + ((long)(n * TT + tz)) * HH * OO);
        const float4* Wg = (const float4*)(Wt + k * OO * OO);
        for (int i = tid; i < 1024; i += 128) {
            float4 v = Zg[i];
            int r = i >> 4;
            int c = (i & 15) * 4;
            float* p = &Zs[r * LDSTRIDE + c];
            p[0] = v.x; p[1] = v.y; p[2] = v.z; p[3] = v.w;
            float4 w = Wg[i];
            float* q = &Wk[r * LDSTRIDE + c];
            q[0] = w.x; q[1] = w.y; q[2] = w.z; q[3] = w.w;
        }
        __syncthreads();

        const int arow = (16 * wave + l15) * LDSTRIDE;  // A row = output channel o
        #pragma unroll
        for (int kk = 0; kk < OO; kk += 4) {
            const int ka = kk + 2 * hi;
            v2f a = *(const v2f*)&Wk[arow + ka];
            #pragma unroll
            for (int j = 0; j < 4; ++j) {
                // B[o'][h] = z[h][o']: contiguous K pair in natural layout
                v2f b = *(const v2f*)&Zs[(16 * j + l15) * LDSTRIDE + ka];
                acc[j] = __builtin_amdgcn_wmma_f32_16x16x4_f32(
                    false, a, false, b, (short)0, acc[j], false, false);
            }
        }
    }

    // epilogue: +b_tcn, BN2 -> stage D into Zs[h][o]
    __syncthreads();   // all waves done reading Zs/Wk
    #pragma unroll
    for (int r = 0; r < 8; ++r) {
        int o = 16 * wave + 8 * hi + r;
        float bb = btcn[o];
        float iv = g2[o] * BN_INV;
        float bt = be2[o];
        #pragma unroll
        for (int j = 0; j < 4; ++j) {
            int h = 16 * j + l15;
            float v = acc[j][r] + bb;
            Zs[h * LDSTRIDE + o] = v * iv + bt;
        }
    }
    __syncthreads();

    // coalesced copy: + residual X, ReLU, float4 out
    const long base = ((long)(n * TT + t)) * HH * OO;
    const float4* Xg4 = (const float4*)(X + base);
    float4* Og4 = (float4*)(out + base);
    for (int i = tid; i < 1024; i += 128) {
        int h = i >> 4;
        int c = (i & 15) * 4;
        const float* p = &Zs[h * LDSTRIDE + c];
        float4 xv = Xg4[i];
        float4 v;
        v.x = fmaxf(p[0] + xv.x, 0.f);
        v.y = fmaxf(p[1] + xv.y, 0.f);
        v.z = fmaxf(p[2] + xv.z, 0.f);
        v.w = fmaxf(p[3] + xv.w, 0.f);
        Og4[i] = v;
    }
}

// ---------------- launcher ----------------

extern "C" void kernel_launch(void* const* d_in, const int* in_sizes, int n_in,
                              void* d_out, int out_size, void* d_ws, size_t ws_size,
                              hipStream_t stream) {
    const float* X      = (const float*)d_in[0];
    const int*   edges  = (const int*)d_in[1];
    const float* w_att1 = (const float*)d_in[2];
    const float* b_att1 = (const float*)d_in[3];
    const float* w_att2 = (const float*)d_in[4];
    const float* b_att2 = (const float*)d_in[5];
    const float* W_lin  = (const float*)d_in[6];
    const float* b_lin  = (const float*)d_in[7];
    const float* bn1g   = (const float*)d_in[8];
    const float* bn1b   = (const float*)d_in[9];
    const float* W_tcn  = (const float*)d_in[10];
    const float* b_tcn  = (const float*)d_in[11];
    const float* bn2g   = (const float*)d_in[12];
    const float* bn2b   = (const float*)d_in[13];

    float* ws   = (float*)d_ws;
    float* adj  = ws + ADJ_OFF;
    float* diag = ws + DIAG_OFF;
    float* Wt   = ws + WT_OFF;
    float* z    = ws + Z_OFF;
    float* out  = (float*)d_out;

    k_zero_adj<<<(NB * HH * HH + 255) / 256, 256, 0, stream>>>(adj);
    k_scatter<<<(NB * EE + 255) / 256, 256, 0, stream>>>(edges, adj);
    k_wtrans<<<(KWIN * OO * OO + 255) / 256, 256, 0, stream>>>(W_tcn, Wt);
    k_att_diag<<<NB * TT, 64, 0, stream>>>(X, w_att1, b_att1, w_att2, b_att2, adj, diag);
    k_support<<<(NB * TT * HH) / 64, 128, 0, stream>>>(X, W_lin, b_lin, bn1g, bn1b, diag, z);
    k_conv<<<NB * TT, 128, 0, stream>>>(z, Wt, b_tcn, bn2g, bn2b, X, out);
}